// CubePadding_15247133901183
// MI455X (gfx1250) — compile-verified
//
#include <hip/hip_runtime.h>
#include <hip/hip_bf16.h>

// Cube padding: x[6, 128, 512, 512] f32, pad p -> out[6, 128, 512+2p, 512+2p]
// Pure streaming op: ~1.6 GB HBM traffic, 0 FLOPs -> HBM-bandwidth bound
// (~70 us at 23.3 TB/s). Strategy: coalesced dword-per-lane (wave32 x 4B =
// one 128B cacheline per vmem instruction) with gfx1250 non-temporal cache
// hints on the streamed interior; branchy gather only on the thin halo.
// HO is a compile-time template parameter so t/HO lowers to mulhi+shift
// instead of the runtime reciprocal-division sequence.

#define FACES 6
#define CH    128
#define SSZ   512   // face height == width

template <int HO_CT>
__global__ __launch_bounds__(256) void cubepad_kernel(
    const float* __restrict__ x, float* __restrict__ out,
    const int p, const int HoRt) {

  const int Ho = (HO_CT > 0) ? HO_CT : HoRt;    // compile-time when HO_CT>0
  const int plane = blockIdx.y;                 // f*CH + c in [0, 768)
  const int t = blockIdx.x * 256 + threadIdx.x; // linear index in output plane
  const int planeElems = Ho * Ho;
  if (t >= planeElems) return;

  const int i = t / Ho;                         // const-div -> mulhi+shift
  const int j = t - i * Ho;
  const int f = plane / CH;                     // 0=back,1=down,2=front,3=left,4=right,5=top
  const int c = plane - f * CH;

  const int inPlane = plane * (SSZ * SSZ);
  float v;

  if (__builtin_expect(i >= p && i < p + SSZ && j >= p && j < p + SSZ, 1)) {
    // ---- interior fast path: straight streaming copy (NT load) ----
    v = __builtin_nontemporal_load(x + inPlane + (i - p) * SSZ + (j - p));
  } else {
    // ---- halo gather: resolve (srcFace, row, col) per reference algebra ----
    int sf, sr, sc;
    if (i < p) {
      // top strip t[f][c][a][b]; corners tile t[a][0] / t[a][S-1] -> clamp b
      const int a = i;
      int b = j - p; b = b < 0 ? 0 : (b > SSZ - 1 ? SSZ - 1 : b);
      switch (f) {
        case 0:  sf = 5; sr = a;            sc = SSZ - 1 - b;  break; // ft[:p,:][::-1 cols]
        case 1:  sf = 2; sr = SSZ - p + a;  sc = b;            break; // ff[-p:,:]
        case 2:  sf = 5; sr = SSZ - p + a;  sc = b;            break; // ft[-p:,:]
        case 3:  sf = 5; sr = b;            sc = a;            break; // swap(ft[:,:p])
        case 4:  sf = 5; sr = SSZ - 1 - b;  sc = SSZ - p + a;  break; // swap(ft[:,-p:])[::-1]
        default: sf = 0; sr = a;            sc = SSZ - 1 - b;  break; // fb[:p,:][::-1 cols]
      }
    } else if (i >= p + SSZ) {
      // down strip d[f][c][a][b]; corners clamp b
      const int a = i - (p + SSZ);
      int b = j - p; b = b < 0 ? 0 : (b > SSZ - 1 ? SSZ - 1 : b);
      switch (f) {
        case 0:  sf = 1; sr = SSZ - p + a;  sc = SSZ - 1 - b;  break; // fd[-p:,:][::-1 cols]
        case 1:  sf = 0; sr = SSZ - p + a;  sc = SSZ - 1 - b;  break; // fb[-p:,:][::-1 cols]
        case 2:  sf = 1; sr = a;            sc = b;            break; // fd[:p,:]
        case 3:  sf = 1; sr = SSZ - 1 - b;  sc = a;            break; // swap(fd[:,:p])[::-1]
        case 4:  sf = 1; sr = b;            sc = SSZ - p + a;  break; // swap(fd[:,-p:])
        default: sf = 2; sr = a;            sc = b;            break; // ff[:p,:]
      }
    } else if (j < p) {
      // left strip l[f][c][a][b]
      const int a = i - p;
      const int b = j;
      switch (f) {
        case 0:  sf = 4; sr = a;            sc = SSZ - p + b;  break; // fr[:,-p:]
        case 1:  sf = 3; sr = SSZ - p + b;  sc = SSZ - 1 - a;  break; // swap(fl[-p:,:])[rev rows]
        case 2:  sf = 3; sr = a;            sc = SSZ - p + b;  break; // fl[:,-p:]
        case 3:  sf = 0; sr = a;            sc = SSZ - p + b;  break; // fb[:,-p:]
        case 4:  sf = 2; sr = a;            sc = SSZ - p + b;  break; // ff[:,-p:]
        default: sf = 3; sr = b;            sc = a;            break; // swap(fl[:p,:])
      }
    } else {
      // right strip r[f][c][a][b]
      const int a = i - p;
      const int b = j - (p + SSZ);
      switch (f) {
        case 0:  sf = 3; sr = a;            sc = b;            break; // fl[:,:p]
        case 1:  sf = 4; sr = SSZ - p + b;  sc = a;            break; // swap(fr[-p:,:])
        case 2:  sf = 4; sr = a;            sc = b;            break; // fr[:,:p]
        case 3:  sf = 2; sr = a;            sc = b;            break; // ff[:,:p]
        case 4:  sf = 0; sr = a;            sc = b;            break; // fb[:,:p]
        default: sf = 4; sr = b;            sc = SSZ - 1 - a;  break; // swap(fr[:p,:])[rev rows]
      }
    }
    v = x[(sf * CH + c) * (SSZ * SSZ) + sr * SSZ + sc];
  }

  // streamed output, written once, never re-read -> non-temporal store
  __builtin_nontemporal_store(v, out + plane * planeElems + t);
}

extern "C" void kernel_launch(void* const* d_in, const int* in_sizes, int n_in,
                              void* d_out, int out_size, void* d_ws, size_t ws_size,
                              hipStream_t stream) {
  const float* x = (const float*)d_in[0];
  float* out = (float*)d_out;

  // Derive padded side length and pad from output size (6*128 planes of Ho*Ho).
  const int planeElems = out_size / (FACES * CH);
  int Ho = SSZ;                       // Ho = SSZ + 2p, integer sqrt via scan
  while (Ho * Ho < planeElems) ++Ho;
  const int p = (Ho - SSZ) / 2;

  dim3 block(256, 1, 1);
  dim3 grid((unsigned)((planeElems + 255) / 256), FACES * CH, 1);

  if (Ho == SSZ + 2) {
    // p == 1 (harness config): fully constant-folded index math
    cubepad_kernel<SSZ + 2><<<grid, block, 0, stream>>>(x, out, p, Ho);
  } else {
    cubepad_kernel<0><<<grid, block, 0, stream>>>(x, out, p, Ho);
  }
}